// ICP_33320356282474
// MI455X (gfx1250) — compile-verified
//
#include <hip/hip_runtime.h>

typedef __attribute__((ext_vector_type(2))) float v2f;
typedef __attribute__((ext_vector_type(8))) float v8f;

#define HH 1080
#define WW 1920
#define NPIX (HH * WW)

// Workspace float layout:
//   [0..48]   : 7x7 Gram accumulator of J' = [Jrot(3), Jtrs(3), res]
//   [64..79]  : current 4x4 pose, row-major
//   [80..81]  : (uint bits) global min/max of depth1
//   [128 ...] : normal1 map, float4 per pixel (w unused) -> 16B aligned
#define WS_ACC  0
#define WS_POSE 64
#define WS_MM   80
#define WS_NRM  128

__global__ void icp_init(const float* __restrict__ pose_in, float* __restrict__ ws) {
    int t = threadIdx.x;
    if (t < 49) ws[WS_ACC + t] = 0.0f;
    if (t < 16) ws[WS_POSE + t] = pose_in[t];
    if (t == 0) {
        unsigned* mm = (unsigned*)(ws + WS_MM);
        mm[0] = 0x7f800000u;  // +inf
        mm[1] = 0u;           // 0.0f (depths are > 0)
    }
}

__global__ void icp_minmax(const float* __restrict__ depth1, float* __restrict__ ws) {
    unsigned* mm = (unsigned*)(ws + WS_MM);
    int tid = blockIdx.x * blockDim.x + threadIdx.x;
    int stride = gridDim.x * blockDim.x;
    float mn = 1e30f, mx = -1e30f;
    for (int i = tid; i < NPIX; i += stride) {
        float d = depth1[i];
        mn = fminf(mn, d);
        mx = fmaxf(mx, d);
    }
#pragma unroll
    for (int off = 16; off > 0; off >>= 1) {
        mn = fminf(mn, __shfl_xor(mn, off, 32));
        mx = fmaxf(mx, __shfl_xor(mx, off, 32));
    }
    if ((threadIdx.x & 31) == 0) {
        atomicMin(&mm[0], __float_as_uint(mn));   // positive floats: uint order == float order
        atomicMax(&mm[1], __float_as_uint(mx));
    }
}

// vertex(x,y) = ((x-cx)/fx*d, (y-cy)/fy*d, d)
__device__ __forceinline__ void vert_at(const float* __restrict__ depth, int x, int y,
                                        float fx, float fy, float cx, float cy,
                                        float& vx, float& vy, float& vz) {
    float d = depth[y * WW + x];
    vx = ((float)x - cx) / fx * d;
    vy = ((float)y - cy) / fy * d;
    vz = d;
}

__global__ void __launch_bounds__(256) icp_normal(const float* __restrict__ depth1,
                                                  const float* __restrict__ Kmat,
                                                  float* __restrict__ ws) {
    int p = blockIdx.x * blockDim.x + threadIdx.x;
    if (p >= NPIX) return;
    float fx = Kmat[0], cx = Kmat[2], fy = Kmat[4], cy = Kmat[5];
    const unsigned* mm = (const unsigned*)(ws + WS_MM);
    float dmin = __uint_as_float(mm[0]);
    float dmax = __uint_as_float(mm[1]);

    int y = p / WW;
    int x = p - y * WW;
    int xm = max(x - 1, 0), xp = min(x + 1, WW - 1);
    int ym = max(y - 1, 0), yp = min(y + 1, HH - 1);

    float tlx, tly, tlz, tx, ty, tz, trx, try_, trz;
    float lx, ly, lz, rx, ry, rz;
    float blx, bly, blz, bx, by, bz, brx, bry, brz;
    vert_at(depth1, xm, ym, fx, fy, cx, cy, tlx, tly, tlz);
    vert_at(depth1, x,  ym, fx, fy, cx, cy, tx,  ty,  tz);
    vert_at(depth1, xp, ym, fx, fy, cx, cy, trx, try_, trz);
    vert_at(depth1, xm, y,  fx, fy, cx, cy, lx,  ly,  lz);
    vert_at(depth1, xp, y,  fx, fy, cx, cy, rx,  ry,  rz);
    vert_at(depth1, xm, yp, fx, fy, cx, cy, blx, bly, blz);
    vert_at(depth1, x,  yp, fx, fy, cx, cy, bx,  by,  bz);
    vert_at(depth1, xp, yp, fx, fy, cx, cy, brx, bry, brz);

    // Sobel: dx = -TL + TR - 2L + 2R - BL + BR ; dy = -TL - 2T - TR + BL + 2B + BR
    float dxx = -tlx + trx - 2.0f * lx + 2.0f * rx - blx + brx;
    float dxy = -tly + try_ - 2.0f * ly + 2.0f * ry - bly + bry;
    float dxz = -tlz + trz - 2.0f * lz + 2.0f * rz - blz + brz;
    float dyx = -tlx - 2.0f * tx - trx + blx + 2.0f * bx + brx;
    float dyy = -tly - 2.0f * ty - try_ + bly + 2.0f * by + bry;
    float dyz = -tlz - 2.0f * tz - trz + blz + 2.0f * bz + brz;

    float nx = dxy * dyz - dxz * dyy;
    float ny = dxz * dyx - dxx * dyz;
    float nz = dxx * dyy - dxy * dyx;
    float inv = 1.0f / (sqrtf(nx * nx + ny * ny + nz * nz) + 1e-8f);
    nx *= inv; ny *= inv; nz *= inv;

    float d = depth1[p];
    bool invalid = (d <= dmin) || (d >= dmax);
    float4* nrm = (float4*)(ws + WS_NRM);
    nrm[p] = invalid ? make_float4(0.f, 0.f, 0.f, 0.f) : make_float4(nx, ny, nz, 0.f);
}

// Gram accumulation: each wave stages 32 pixels' J' (padded to 16 floats) in LDS,
// then 8x V_WMMA_F32_16X16X4_F32 accumulate C += A * A^T (same fragment as A & B:
// A-layout (16x4: lanes0-15 K0/K1, lanes16-31 K2/K3) and B-layout (4x16: row=v+2*half)
// address identical LDS elements, so one v2f serves both operands).
#define ACC_WGS 256
#define ACC_WPW 8
#define ACC_NWG 900   // 900*8 = 7200 waves; 64800 chunks / 7200 = 9 chunks per wave

__global__ void __launch_bounds__(256) icp_accum(const float* __restrict__ depth0,
                                                 const float* __restrict__ depth1,
                                                 const float* __restrict__ Kmat,
                                                 float* __restrict__ ws) {
    __shared__ __align__(16) float stage[ACC_WPW][32 * 16];
    __shared__ float sacc[49];
    int lane = threadIdx.x & 31;
    int wv = threadIdx.x >> 5;
    if (threadIdx.x < 49) sacc[threadIdx.x] = 0.0f;
    __syncthreads();

    float fx = Kmat[0], cx = Kmat[2], fy = Kmat[4], cy = Kmat[5];
    const float* P = ws + WS_POSE;
    float R00 = P[0], R01 = P[1], R02 = P[2],  T0 = P[3];
    float R10 = P[4], R11 = P[5], R12 = P[6],  T1 = P[7];
    float R20 = P[8], R21 = P[9], R22 = P[10], T2 = P[11];
    const float4* nrm = (const float4*)(ws + WS_NRM);

    int gwave = blockIdx.x * ACC_WPW + wv;
    int nwaves = gridDim.x * ACC_WPW;
    float* st = stage[wv];
    v8f c = {};

    for (int base = gwave * 32; base < NPIX; base += nwaves * 32) {
        int p = base + lane;                      // NPIX % 32 == 0, base % 32 == 0 -> in range
        float d0 = depth0[p];
        int y0i = p / WW;
        int x0i = p - y0i * WW;
        float X = ((float)x0i - cx) / fx * d0;
        float Y = ((float)y0i - cy) / fy * d0;
        float Z = d0;
        float px = R00 * X + R01 * Y + R02 * Z + T0;
        float py = R10 * X + R11 * Y + R12 * Z + T1;
        float pz = R20 * X + R21 * Y + R22 * Z + T2;
        float invz = 1.0f / pz;
        float u = px * invz * fx + cx;
        float v = py * invz * fy + cy;
        bool inview = (u > 0.f) && (u < (float)(WW - 1)) && (v > 0.f) && (v < (float)(HH - 1));
        float uc = fminf(fmaxf(u, 0.f), (float)(WW - 1));
        float vc = fminf(fmaxf(v, 0.f), (float)(HH - 1));
        float x0f = floorf(uc), y0f = floorf(vc);
        int ix0 = (int)x0f, iy0 = (int)y0f;
        int ix1 = min(ix0 + 1, WW - 1), iy1 = min(iy0 + 1, HH - 1);
        float wx = uc - x0f, wy = vc - y0f;
        float w00 = (1.f - wx) * (1.f - wy), w01 = wx * (1.f - wy);
        float w10 = (1.f - wx) * wy,         w11 = wx * wy;

        // bilinear warp of vertex1 (recomputed from depth1 taps)
        float da = depth1[iy0 * WW + ix0], db = depth1[iy0 * WW + ix1];
        float dc = depth1[iy1 * WW + ix0], dd = depth1[iy1 * WW + ix1];
        float ax0 = ((float)ix0 - cx) / fx, ax1 = ((float)ix1 - cx) / fx;
        float ay0 = ((float)iy0 - cy) / fy, ay1 = ((float)iy1 - cy) / fy;
        float rvx = w00 * ax0 * da + w01 * ax1 * db + w10 * ax0 * dc + w11 * ax1 * dd;
        float rvy = w00 * ay0 * da + w01 * ay0 * db + w10 * ay1 * dc + w11 * ay1 * dd;
        float rvz = w00 * da + w01 * db + w10 * dc + w11 * dd;

        // bilinear warp of normal1
        float4 na = nrm[iy0 * WW + ix0], nb = nrm[iy0 * WW + ix1];
        float4 nc = nrm[iy1 * WW + ix0], nd = nrm[iy1 * WW + ix1];
        float rnx = w00 * na.x + w01 * nb.x + w10 * nc.x + w11 * nd.x;
        float rny = w00 * na.y + w01 * nb.y + w10 * nc.y + w11 * nd.y;
        float rnz = w00 * na.z + w01 * nb.z + w10 * nc.z + w11 * nd.z;

        float dfx = px - rvx, dfy = py - rvy, dfz = pz - rvz;
        float dist2 = dfx * dfx + dfy * dfy + dfz * dfz;
        bool occ = (!inview) || (dist2 > 0.01f);                 // 0.1^2
        bool bad = occ || !(d0 > 0.f) || !(rvz > 0.f);
        float g = bad ? 0.f : 1.f;

        float res = rnx * dfx + rny * dfy + rnz * dfz;
        float jr0 = py * rnz - pz * rny;                          // cross(p, rn)
        float jr1 = pz * rnx - px * rnz;
        float jr2 = px * rny - py * rnx;

        float4* s4 = (float4*)&st[lane * 16];
        s4[0] = make_float4(jr0 * g, jr1 * g, jr2 * g, rnx * g);
        s4[1] = make_float4(rny * g, rnz * g, res * g, 0.f);
        s4[2] = make_float4(0.f, 0.f, 0.f, 0.f);
        s4[3] = make_float4(0.f, 0.f, 0.f, 0.f);
        __builtin_amdgcn_wave_barrier();   // wave-synchronous LDS; DS ops are in-order per wave

        int m = lane & 15;
        int kb = (lane >> 4) << 1;         // 0 or 2
#pragma unroll
        for (int kc = 0; kc < 8; ++kc) {
            int pk = kc * 4 + kb;
            v2f a;
            a.x = st[pk * 16 + m];
            a.y = st[(pk + 1) * 16 + m];
            // C(16x16,f32) += A(16x4) * B(4x16); same fragment as both operands => A*A^T
            c = __builtin_amdgcn_wmma_f32_16x16x4_f32(false, a, false, a,
                                                      (short)0, c, false, false);
        }
        __builtin_amdgcn_wave_barrier();
    }

    // C fragment: lane n (<16) vgpr v holds entry (M=v, N=n). Rows/cols 0..6 are live.
    if (lane < 7) {
#pragma unroll
        for (int r = 0; r < 7; ++r) atomicAdd(&sacc[r * 7 + lane], c[r]);   // ds_add_f32
    }
    __syncthreads();
    if (threadIdx.x < 49) atomicAdd(&ws[WS_ACC + threadIdx.x], sacc[threadIdx.x]);
}

__global__ void icp_solve(float* __restrict__ ws, float* __restrict__ out) {
    if (threadIdx.x != 0 || blockIdx.x != 0) return;
    float acc[49];
#pragma unroll
    for (int i = 0; i < 49; ++i) acc[i] = ws[WS_ACC + i];

    float A[6][7];
    float tr = 0.f;
    for (int i = 0; i < 6; ++i) tr += acc[i * 7 + i];
    for (int i = 0; i < 6; ++i) {
        for (int j = 0; j < 6; ++j) A[i][j] = acc[i * 7 + j];
        A[i][i] += tr * 0.001f;
        A[i][6] = -acc[i * 7 + 6];       // reference uses J = -J for JtR
    }
    // Gauss-Jordan (Hess is SPD + damping)
    for (int k = 0; k < 6; ++k) {
        float ip = 1.0f / A[k][k];
        for (int j = k; j < 7; ++j) A[k][j] *= ip;
        for (int i = 0; i < 6; ++i) {
            if (i == k) continue;
            float f = A[i][k];
            for (int j = k; j < 7; ++j) A[i][j] -= f * A[k][j];
        }
    }
    float w0 = A[0][6], w1 = A[1][6], w2 = A[2][6];
    float dt0 = A[3][6], dt1 = A[4][6], dt2 = A[5][6];

    // exp_so3
    float dR[3][3] = {{1, 0, 0}, {0, 1, 0}, {0, 0, 1}};
    float th = sqrtf(w0 * w0 + w1 * w1 + w2 * w2);
    if (th > 1e-10f) {
        float sa = sinf(th) / th;
        float sb = (1.0f - cosf(th)) / (th * th);
        float wh[3][3] = {{0, -w2, w1}, {w2, 0, -w0}, {-w1, w0, 0}};
        float wh2[3][3];
        for (int i = 0; i < 3; ++i)
            for (int j = 0; j < 3; ++j)
                wh2[i][j] = wh[i][0] * wh[0][j] + wh[i][1] * wh[1][j] + wh[i][2] * wh[2][j];
        for (int i = 0; i < 3; ++i)
            for (int j = 0; j < 3; ++j)
                dR[i][j] = (i == j ? 1.f : 0.f) + sa * wh[i][j] + sb * wh2[i][j];
    }

    float* P = ws + WS_POSE;
    float R[3][3], t[3];
    for (int i = 0; i < 3; ++i) {
        for (int j = 0; j < 3; ++j) R[i][j] = P[i * 4 + j];
        t[i] = P[i * 4 + 3];
    }
    float Rn[3][3], tn[3];
    for (int i = 0; i < 3; ++i) {
        for (int j = 0; j < 3; ++j)
            Rn[i][j] = dR[i][0] * R[0][j] + dR[i][1] * R[1][j] + dR[i][2] * R[2][j];
        tn[i] = dR[i][0] * t[0] + dR[i][1] * t[1] + dR[i][2] * t[2] +
                (i == 0 ? dt0 : (i == 1 ? dt1 : dt2));
    }
    for (int i = 0; i < 3; ++i) {
        for (int j = 0; j < 3; ++j) P[i * 4 + j] = Rn[i][j];
        P[i * 4 + 3] = tn[i];
    }
    // publish pose (last iteration's write is the final answer; bottom row is input's 0,0,0,1)
    for (int i = 0; i < 16; ++i) out[i] = P[i];
    // re-zero Gram accumulator for the next iteration / next call
    for (int i = 0; i < 49; ++i) ws[WS_ACC + i] = 0.0f;
}

extern "C" void kernel_launch(void* const* d_in, const int* in_sizes, int n_in,
                              void* d_out, int out_size, void* d_ws, size_t ws_size,
                              hipStream_t stream) {
    const float* pose_in = (const float*)d_in[0];   // 4x4
    const float* depth0  = (const float*)d_in[1];   // 1080x1920
    const float* depth1  = (const float*)d_in[2];   // 1080x1920
    const float* Kmat    = (const float*)d_in[3];   // 3x3
    float* ws  = (float*)d_ws;                      // needs 512B + NPIX*16B (~33.2 MB)
    float* out = (float*)d_out;                     // 16 floats

    icp_init<<<1, 64, 0, stream>>>(pose_in, ws);
    icp_minmax<<<256, 256, 0, stream>>>(depth1, ws);
    icp_normal<<<(NPIX + 255) / 256, 256, 0, stream>>>(depth1, Kmat, ws);
    for (int it = 0; it < 3; ++it) {
        icp_accum<<<ACC_NWG, ACC_WGS, 0, stream>>>(depth0, depth1, Kmat, ws);
        icp_solve<<<1, 32, 0, stream>>>(ws, out);
    }
}